// Router_17025250361618
// MI455X (gfx1250) — compile-verified
//
#include <hip/hip_runtime.h>

typedef float v2f __attribute__((ext_vector_type(2)));
typedef float v4f __attribute__((ext_vector_type(4)));
typedef float v8f __attribute__((ext_vector_type(8)));

#define D_MODEL 1024
#define NUM_EXPERTS 8
#define G_GROUPS 8

// ---------------------------------------------------------------------------
// Kernel 1: gating GEMM (fp32 WMMA 16x16x4) + softmax + top-2 per token.
// One wave = 16 tokens x 16 experts (8 real + 8 zero-padded columns).
// Block = 256 threads = 8 waves = 128 tokens.
// Padding trick: LDS holds 9 weight rows; row 8 is all zeros, and padded
// expert lanes read from it -> no divergent EXEC masking in the K-loop.
// ---------------------------------------------------------------------------
__global__ __launch_bounds__(256) void k_gate(const float* __restrict__ x,
                                              const float* __restrict__ W,
                                              const float* __restrict__ bias,
                                              int* __restrict__ wsIdx,
                                              float* __restrict__ wsGate) {
    __shared__ float wT[(NUM_EXPERTS + 1) * D_MODEL]; // transposed W + zero row, 36 KB
    __shared__ float lg[8][256];                      // per-wave 16x16 logits tile

    const int tid = threadIdx.x;

    // Stage W transposed into LDS: wT[e][d] = W[d*8 + e]; row 8 = zeros.
    for (int i = tid; i < (NUM_EXPERTS + 1) * D_MODEL; i += 256) {
        if (i < NUM_EXPERTS * D_MODEL) {
            int d = i >> 3, e = i & 7;
            wT[e * D_MODEL + d] = W[i];
        } else {
            wT[i] = 0.0f;
        }
    }
    __syncthreads();

    const int wave = tid >> 5;
    const int lane = tid & 31;
    const int half = lane >> 4;      // 0: K pair {0,1}, 1: K pair {2,3}
    const int mrow = lane & 15;      // token row within tile / expert column
    const long tokenBase = ((long)blockIdx.x * 8 + wave) * 16;

    const float* xrow = x + (tokenBase + mrow) * D_MODEL + 2 * half;
    const int brow = (mrow < NUM_EXPERTS) ? mrow : NUM_EXPERTS;  // padded -> zero row
    const float* bptr = &wT[brow * D_MODEL + 2 * half];

    v8f c0 = {}, c1 = {};
    for (int k = 0; k < D_MODEL; k += 8) {
        // A: 16x4 fp32, lanes 0-15 hold K=0,1 ; lanes 16-31 hold K=2,3
        v2f a0 = __builtin_nontemporal_load((const v2f*)(xrow + k));
        v2f b0 = *(const v2f*)(bptr + k);
        c0 = __builtin_amdgcn_wmma_f32_16x16x4_f32(false, a0, false, b0,
                                                   (short)0, c0, false, false);
        v2f a1 = __builtin_nontemporal_load((const v2f*)(xrow + k + 4));
        v2f b1 = *(const v2f*)(bptr + k + 4);
        c1 = __builtin_amdgcn_wmma_f32_16x16x4_f32(false, a1, false, b1,
                                                   (short)0, c1, false, false);
    }
    v8f c = c0 + c1;

    // C/D layout: VGPR j holds M=j (lanes 0-15) / M=j+8 (lanes 16-31), N=lane&15
#pragma unroll
    for (int j = 0; j < 8; ++j)
        lg[wave][(j + 8 * half) * 16 + mrow] = c[j];
    __syncthreads();

    if (lane < 16) {
        const long t = tokenBase + lane;
        float l[NUM_EXPERTS];
        float m = -1e30f;
#pragma unroll
        for (int e = 0; e < NUM_EXPERTS; ++e) {
            l[e] = lg[wave][lane * 16 + e] + bias[e];
            m = fmaxf(m, l[e]);
        }
        float s = 0.0f;
#pragma unroll
        for (int e = 0; e < NUM_EXPERTS; ++e) { l[e] = __expf(l[e] - m); s += l[e]; }
        const float inv = 1.0f / s;
        int i0 = 0; float p0 = -1.0f;
#pragma unroll
        for (int e = 0; e < NUM_EXPERTS; ++e) {
            float p = l[e] * inv; l[e] = p;
            if (p > p0) { p0 = p; i0 = e; }     // strict > : lowest index wins ties
        }
        int i1 = 0; float p1 = -1.0f;
#pragma unroll
        for (int e = 0; e < NUM_EXPERTS; ++e) {
            if (e != i0 && l[e] > p1) { p1 = l[e]; i1 = e; }
        }
        wsIdx[2 * t]     = i0;  wsIdx[2 * t + 1]  = i1;
        wsGate[2 * t]    = p0;  wsGate[2 * t + 1] = p1;
    }
}

// ---------------------------------------------------------------------------
// Kernel 2: per-(group, slot, expert) cumulative position over S tokens.
// Block = 64 threads = 2 waves (wave k = top-k slot k). Each lane scans a
// contiguous chunk of S/32 tokens: local count -> cross-lane exclusive
// prefix via LDS -> second pass emits 1-based positions.
// ---------------------------------------------------------------------------
__global__ __launch_bounds__(64) void k_scan(const int* __restrict__ wsIdx,
                                             int* __restrict__ wsPos, int S) {
    __shared__ int cnts[2][32][NUM_EXPERTS];
    const int g = blockIdx.x;
    const int k = threadIdx.x >> 5;
    const int lane = threadIdx.x & 31;
    const int chunk = S >> 5;
    const int base = g * S + lane * chunk;

    int c[NUM_EXPERTS];
#pragma unroll
    for (int e = 0; e < NUM_EXPERTS; ++e) c[e] = 0;
    for (int i = 0; i < chunk; ++i) {
        int idx = wsIdx[2 * (base + i) + k];
#pragma unroll
        for (int e = 0; e < NUM_EXPERTS; ++e) c[e] += (idx == e);
    }
#pragma unroll
    for (int e = 0; e < NUM_EXPERTS; ++e) cnts[k][lane][e] = c[e];
    __syncthreads();

    int run[NUM_EXPERTS];
#pragma unroll
    for (int e = 0; e < NUM_EXPERTS; ++e) run[e] = 0;
    for (int l = 0; l < 32; ++l) {
        if (l < lane) {
#pragma unroll
            for (int e = 0; e < NUM_EXPERTS; ++e) run[e] += cnts[k][l][e];
        }
    }
    for (int i = 0; i < chunk; ++i) {
        int t = base + i;
        int idx = wsIdx[2 * t + k];
        int pos = 0;
#pragma unroll
        for (int e = 0; e < NUM_EXPERTS; ++e) {
            bool m = (idx == e);
            run[e] += m;
            if (m) pos = run[e];
        }
        wsPos[2 * t + k] = pos;   // 1-based position in (slot, expert) stream
    }
}

// ---------------------------------------------------------------------------
// Kernel 3: materialize combine_tensor + dispatch_mask.
// One block per token: zero-fill its E*(cap-1) row with the <=2 scatter
// values patched in-register; non-temporal 128-bit stores (write-once data).
// ---------------------------------------------------------------------------
__device__ __forceinline__ float pickv(int idx, int o0, float g0, int o1, float g1) {
    return idx == o0 ? g0 : (idx == o1 ? g1 : 0.0f);
}

__global__ __launch_bounds__(256) void k_out(const int* __restrict__ wsIdx,
                                             const int* __restrict__ wsPos,
                                             const float* __restrict__ wsGate,
                                             const int* __restrict__ capPtr,
                                             float* __restrict__ out, long N) {
    const long t = blockIdx.x;
    const int cap = *capPtr;
    const int cm1 = cap - 1;
    const int rowlen = NUM_EXPERTS * cm1;          // 2040 for cap=256
    const long cbase = t * (long)rowlen;
    float* comb = out;
    float* disp = out + N;

    const int e0 = wsIdx[2 * t],  e1 = wsIdx[2 * t + 1];
    const int p0 = wsPos[2 * t],  p1 = wsPos[2 * t + 1];
    const float g0 = wsGate[2 * t], g1 = wsGate[2 * t + 1];
    const float d0 = (g0 != 0.0f) ? 1.0f : 0.0f;
    const float d1 = (g1 != 0.0f) ? 1.0f : 0.0f;
    const int off0 = (p0 >= 1 && p0 < cap) ? (e0 * cm1 + p0 - 1) : -1;
    const int off1 = (p1 >= 1 && p1 < cap) ? (e1 * cm1 + p1 - 1) : -1;

    if ((rowlen & 3) == 0) {
        const int nv4 = rowlen >> 2;
        v4f* cdst = (v4f*)(comb + cbase);
        v4f* ddst = (v4f*)(disp + cbase);
        for (int i = threadIdx.x; i < nv4; i += 256) {
            const int b = i << 2;
            v4f cv, dv;
#pragma unroll
            for (int j = 0; j < 4; ++j) {
                cv[j] = pickv(b + j, off0, g0, off1, g1);
                dv[j] = pickv(b + j, off0, d0, off1, d1);
            }
            __builtin_nontemporal_store(cv, cdst + i);
            __builtin_nontemporal_store(dv, ddst + i);
        }
    } else {
        for (int i = threadIdx.x; i < rowlen; i += 256) {
            comb[cbase + i] = pickv(i, off0, g0, off1, g1);
            disp[cbase + i] = pickv(i, off0, d0, off1, d1);
        }
    }
}

// ---------------------------------------------------------------------------
extern "C" void kernel_launch(void* const* d_in, const int* in_sizes, int n_in,
                              void* d_out, int out_size, void* d_ws, size_t ws_size,
                              hipStream_t stream) {
    const float* x    = (const float*)d_in[0];
    const float* W    = (const float*)d_in[1];
    const float* bias = (const float*)d_in[2];
    const int*   cap  = (const int*)d_in[3];

    const long T = (long)in_sizes[0] / D_MODEL;   // G*S = 16384 tokens
    const int  S = (int)(T / G_GROUPS);           // 2048

    // Workspace: [2T ints idx][2T ints pos][2T floats gate]
    int*   wsIdx  = (int*)d_ws;
    int*   wsPos  = wsIdx + 2 * T;
    float* wsGate = (float*)(wsPos + 2 * T);

    float* out = (float*)d_out;
    const long N = (long)out_size / 2;            // combine | dispatch halves

    k_gate<<<dim3((unsigned)(T / 128)), dim3(256), 0, stream>>>(x, W, bias, wsIdx, wsGate);
    k_scan<<<dim3(G_GROUPS), dim3(64), 0, stream>>>(wsIdx, wsPos, S);
    k_out <<<dim3((unsigned)T), dim3(256), 0, stream>>>(wsIdx, wsPos, wsGate, cap, out, N);
}